// VideoRandomPerspective_28638841930502
// MI455X (gfx1250) — compile-verified
//
#include <hip/hip_runtime.h>
#include <math.h>

typedef __attribute__((ext_vector_type(2))) float v2f;
typedef __attribute__((ext_vector_type(4))) float v4f;
typedef __attribute__((ext_vector_type(8))) float v8f;

#define IMG_H 224
#define IMG_W 224
#define NIMG  256           // B*F
#define NROW  766           // 3*L - 2
#define NTAP  511           // 2*L - 1

// -------------------------------------------------------------------------
// Kernel 1: temporal smoothing (Toeplitz matmul via V_WMMA_F32_16X16X4_F32)
// + per-image 3x3 homography chain. One block, 8 waves (wave32).
//
// vals[r][i] = sum_k noise[r][i+k] * w[k],  k in [0,511)
// Tiled as 64 WMMA tiles (4 rows x 16 groups of 16 outputs), K accumulated
// 4 taps per WMMA over 512 padded taps (w[511] = 0).
//   A (16x4): A[m][j] = noise[r][g*16 + m + k0 + j]
//   B (4x16): B[j][n] = w[k0 + j]   (all 16 columns identical)
// ISA f32 16x4 layout: lanes 0-15 hold K=0,1 ; lanes 16-31 hold K=2,3.
// D 16x16 f32: VGPR d, lanes 0-15 = row d, lanes 16-31 = row 8+d; all
// columns equal, so lane 0 / lane 16 carry the result.
// -------------------------------------------------------------------------
__global__ __launch_bounds__(256) void vrp_matrices(
    const float* __restrict__ noise,   // 4 x 766
    const float* __restrict__ basis,   // 4 x 3 x 3
    float* __restrict__ Mout)          // 256 x 9
{
  __shared__ float sNoise[4 * 768];    // padded rows, tail zeroed
  __shared__ float sW[512];            // padded window, sW[511] = 0
  __shared__ float sVals[4 * 256];

  const int tid = threadIdx.x;

  // Stage noise rows into LDS (zero-pad tail so pad-tap reads are 0, not NaN)
  for (int i = tid; i < 4 * 768; i += 256) {
    const int r = i / 768;
    const int k = i - r * 768;
    sNoise[i] = (k < NROW) ? noise[r * NROW + k] : 0.0f;
  }

  // Window: w[k] = 1.1^(255 - |k-255|) / S,  S = sum of all 511 taps
  const float p255 = powf(1.1f, 255.0f);
  const float S = (p255 - 1.0f) * 10.0f + (p255 * 1.1f - 1.0f) * 10.0f;
  for (int k = tid; k < 512; k += 256) {
    float w = 0.0f;
    if (k < NTAP) {
      const float e = 255.0f - fabsf((float)k - 255.0f);
      w = powf(1.1f, e) / S;
    }
    sW[k] = w;
  }
  __syncthreads();

  const int wave = tid >> 5;           // 8 waves of 32
  const int lane = tid & 31;
  const int m    = lane & 15;
  const int koff = (lane < 16) ? 0 : 2;

  for (int i = 0; i < 8; ++i) {
    const int t = wave * 8 + i;        // tile 0..63
    const int r = t >> 4;              // noise row 0..3
    const int g = t & 15;              // output group of 16
    const float* nrow = &sNoise[r * 768 + g * 16 + m];
    v8f acc = {};
    for (int k0 = 0; k0 < 512; k0 += 4) {
      v2f a, b;
      a.x = nrow[k0 + koff];
      a.y = nrow[k0 + koff + 1];
      b.x = sW[k0 + koff];
      b.y = sW[k0 + koff + 1];
      acc = __builtin_amdgcn_wmma_f32_16x16x4_f32(
          /*neg_a=*/false, a, /*neg_b=*/false, b,
          /*c_mod=*/(short)0, acc, /*reuse_a=*/false, /*reuse_b=*/false);
    }
    if (lane == 0 || lane == 16) {
      const int mbase = g * 16 + ((lane == 16) ? 8 : 0);
#pragma unroll
      for (int d = 0; d < 8; ++d) sVals[r * 256 + mbase + d] = acc[d];
    }
  }
  __syncthreads();

  // Per-image matrix chain: one image per thread.
  {
    const int l = tid;
    const float wv0 = fmaxf(sVals[0 * 256 + l], 0.0f);
    const float wv1 = fmaxf(sVals[1 * 256 + l], 0.0f);
    const float wv2 = fmaxf(sVals[2 * 256 + l], 0.0f);
    const float wv3 = fmaxf(sVals[3 * 256 + l], 0.0f);
    const float rem = 4.0f - (wv0 + wv1 + wv2 + wv3);

    float Wm[9];
#pragma unroll
    for (int j = 0; j < 9; ++j) {
      const float idv = (j == 0 || j == 4 || j == 8) ? 1.0f : 0.0f;
      Wm[j] = wv0 * basis[0 * 9 + j] + wv1 * basis[1 * 9 + j] +
              wv2 * basis[2 * 9 + j] + wv3 * basis[3 * 9 + j] + rem * idv;
    }

    // T = Wm @ Ainv,  Ainv = [[1/223,0,-0.5],[0,1/223,-0.5],[0,0,1]]
    const float s = 1.0f / 223.0f;
    float T[9];
#pragma unroll
    for (int i2 = 0; i2 < 3; ++i2) {
      T[i2 * 3 + 0] = Wm[i2 * 3 + 0] * s;
      T[i2 * 3 + 1] = Wm[i2 * 3 + 1] * s;
      T[i2 * 3 + 2] = -0.5f * Wm[i2 * 3 + 0] - 0.5f * Wm[i2 * 3 + 1] + Wm[i2 * 3 + 2];
    }
    // M = A @ T,  A = [[223,0,111.5],[0,223,111.5],[0,0,1]]
    float M[9];
#pragma unroll
    for (int j = 0; j < 3; ++j) {
      M[0 * 3 + j] = 223.0f * T[0 * 3 + j] + 111.5f * T[2 * 3 + j];
      M[1 * 3 + j] = 223.0f * T[1 * 3 + j] + 111.5f * T[2 * 3 + j];
      M[2 * 3 + j] = T[2 * 3 + j];
    }
#pragma unroll
    for (int j = 0; j < 9; ++j) Mout[l * 9 + j] = M[j];
  }
}

// -------------------------------------------------------------------------
// Kernel 2: projective warp + bilinear gather. 4 x-pixels per thread,
// 3 aligned 16B non-temporal stores per thread (output never re-read;
// keep the 154MB input resident in the 192MB L2).
// 49 blocks per image (224*56 quads / 256 threads) -> M is block-uniform.
// -------------------------------------------------------------------------
__global__ __launch_bounds__(256) void vrp_warp(
    const float* __restrict__ x,       // 256 x 224 x 224 x 3
    const float* __restrict__ Mmat,    // 256 x 9
    float* __restrict__ out)           // 256 x 224 x 224 x 3
{
  __shared__ float sM[9];
  const int tid = threadIdx.x;
  const int l = blockIdx.x / 49;
  if (tid < 9) sM[tid] = Mmat[l * 9 + tid];
  __syncthreads();

  const int q  = (blockIdx.x % 49) * 256 + tid;  // quad id 0..12543
  const int y  = q / 56;
  const int x0 = (q - y * 56) * 4;

  const float m0 = sM[0], m1 = sM[1], m2 = sM[2];
  const float m3 = sM[3], m4 = sM[4], m5 = sM[5];
  const float m6 = sM[6], m7 = sM[7], m8 = sM[8];

  const float* __restrict__ img = x + (size_t)l * (IMG_H * IMG_W * 3);
  const float fy = (float)y;
  float res[12];

#pragma unroll
  for (int i = 0; i < 4; ++i) {
    const float fx = (float)(x0 + i);
    const float pw = m6 * fx + m7 * fy + m8;
    const float inv = 1.0f / pw;
    const float sx = (m0 * fx + m1 * fy + m2) * inv;
    const float sy = (m3 * fx + m4 * fy + m5) * inv;
    const float xf = floorf(sx);
    const float yf = floorf(sy);
    const float wx = sx - xf;
    const float wy = sy - yf;
    const int xi = (int)xf;
    const int yi = (int)yf;

    float c0 = 0.0f, c1 = 0.0f, c2 = 0.0f;
#pragma unroll
    for (int dy = 0; dy < 2; ++dy) {
#pragma unroll
      for (int dx = 0; dx < 2; ++dx) {
        const int tx = xi + dx;
        const int ty = yi + dy;
        const bool v = (tx >= 0) && (tx < IMG_W) && (ty >= 0) && (ty < IMG_H);
        const float wtap = (dy ? wy : 1.0f - wy) * (dx ? wx : 1.0f - wx);
        const float wgt = v ? wtap : 0.0f;
        const int cx = min(max(tx, 0), IMG_W - 1);
        const int cy = min(max(ty, 0), IMG_H - 1);
        const float* p = img + (cy * IMG_W + cx) * 3;
        c0 = fmaf(p[0], wgt, c0);
        c1 = fmaf(p[1], wgt, c1);
        c2 = fmaf(p[2], wgt, c2);
      }
    }
    res[i * 3 + 0] = c0;
    res[i * 3 + 1] = c1;
    res[i * 3 + 2] = c2;
  }

  // 12 floats = 48 bytes, base offset 48*t -> 16B aligned b128 stores
  v4f* o4 = (v4f*)(out + ((size_t)l * (IMG_H * IMG_W) + (size_t)y * IMG_W + x0) * 3);
  v4f v0 = {res[0], res[1], res[2], res[3]};
  v4f v1 = {res[4], res[5], res[6], res[7]};
  v4f v2 = {res[8], res[9], res[10], res[11]};
  __builtin_nontemporal_store(v0, o4 + 0);
  __builtin_nontemporal_store(v1, o4 + 1);
  __builtin_nontemporal_store(v2, o4 + 2);
}

extern "C" void kernel_launch(void* const* d_in, const int* in_sizes, int n_in,
                              void* d_out, int out_size, void* d_ws, size_t ws_size,
                              hipStream_t stream) {
  const float* x     = (const float*)d_in[0];  // (8,32,224,224,3) f32
  const float* noise = (const float*)d_in[1];  // (4,766) f32
  const float* basis = (const float*)d_in[2];  // (4,3,3) f32
  float* out = (float*)d_out;
  float* Mws = (float*)d_ws;                   // 256*9 floats of scratch

  (void)in_sizes; (void)n_in; (void)out_size; (void)ws_size;

  vrp_matrices<<<1, 256, 0, stream>>>(noise, basis, Mws);
  vrp_warp<<<(NIMG * 49), 256, 0, stream>>>(x, Mws, out);
}